// FastDiceLoss_16200616640591
// MI455X (gfx1250) — compile-verified
//
#include <hip/hip_runtime.h>
#include <hip/hip_bf16.h>
#include <stddef.h>

// ---------------- problem constants (from reference) ----------------
#define NIMG 8
#define CFEAT 8
#define HH 128
#define WW 128
#define HWSZ (HH * WW)          // 16384
#define KOBJ 32
#define PPAR 169
#define NPAIR (NIMG * KOBJ / 2) // 128 instance pairs (2 instances share a WMMA)
#define PAIR_STRIDE 512         // floats per pair in workspace
#define ACC_BASE (NPAIR * PAIR_STRIDE) // float offset of per-block partials

typedef float v2f __attribute__((ext_vector_type(2)));
typedef float v8f __attribute__((ext_vector_type(8)));

// Orientation: D = A x B with A = weights (M = 16 output channels for the
// 2 packed instances, K = input channels) and B = data (K = input channels,
// N = 16 pixels).  D then has channels in VGPRs and pixels in lanes, which is
// the *B-operand* layout of the next layer up to a half-wave swap ->
// v_permlane16_swap chains layers with 4 VALU ops instead of an LDS transpose.
//
// Workspace per pair (floats), fragment-value layout identical for A-use:
//   [0,192)   W0 fragments: ((kg*2+hi)*16+lane16)*2+s -> w0[lane16][4kg+2hi+s]
//             (K=12 features: 8 seg + x + y + bias-const + pad)
//   [192,448) W1 fragments: same scheme, kg<4 (K=16 block-diagonal)
//   [448,464) b1 per channel (C-init for layer 1, indexed by v+8*hi)
//   [464,480) w2: [w2_inst0(8) | w2_inst1(8)]
//   [480,482) b2 per instance
// ---------------------------------------------------------------------

__global__ void prep_kernel(const float* __restrict__ cw,
                            const int* __restrict__ ind,
                            float* __restrict__ ws) {
  __shared__ float sp[2][PPAR];
  __shared__ int sidx[2];
  const int pair = blockIdx.x;
  const int n = pair >> 4;
  const int ip = pair & 15;

  if (threadIdx.x < 2) sidx[threadIdx.x] = ind[n * KOBJ + ip * 2 + threadIdx.x];
  __syncthreads();

  // gather 2x169 dynamic params (64KB-strided gather into conv_weight)
  for (int t = threadIdx.x; t < 2 * PPAR; t += blockDim.x) {
    const int i = t / PPAR, q = t - i * PPAR;
    sp[i][q] = cw[(size_t)(n * PPAR + q) * HWSZ + sidx[i]];
  }
  __syncthreads();

  float* wp = ws + (size_t)pair * PAIR_STRIDE;

  // W0 fragments (layer 0), xrel/yrel folded: feature = [seg0..7, x, y, 1, 0]
  for (int e = threadIdx.x; e < 192; e += blockDim.x) {
    const int s = e & 1;
    const int col = (e >> 1) & 15;   // output channel (M)
    const int gh = e >> 5;           // kg*2+hi in [0,6)
    const int hi = gh & 1, kg = gh >> 1;
    const int row = 4 * kg + 2 * hi + s;   // K-channel 0..11
    const int i = col >> 3, o = col & 7;
    float v;
    if (row < 8) {
      v = sp[i][o * 10 + row];                       // w0[o][c]
    } else if (row == 8) {
      v = sp[i][o * 10 + 8] * (1.0f / 128.0f);       // x-grid weight
    } else if (row == 9) {
      v = sp[i][o * 10 + 9] * (1.0f / 128.0f);       // y-grid weight
    } else if (row == 10) {                          // adjusted bias b0'
      const float xm = (float)(sidx[i] % WW);
      const float ym = (float)(sidx[i] / WW);
      v = sp[i][152 + o] -
          (sp[i][o * 10 + 8] * xm + sp[i][o * 10 + 9] * ym) * (1.0f / 128.0f);
    } else {
      v = 0.0f;                                      // pad channel
    }
    wp[e] = v;
  }

  // W1 fragments (layer 1), block-diagonal over the two instances
  for (int e = threadIdx.x; e < 256; e += blockDim.x) {
    const int s = e & 1;
    const int col = (e >> 1) & 15;
    const int gh = e >> 5;
    const int hi = gh & 1, kg = gh >> 1;
    const int row = 4 * kg + 2 * hi + s;   // K 0..15
    const int i = col >> 3, o = col & 7;
    float v = 0.0f;
    if ((row >> 3) == i) v = sp[i][80 + o * 8 + (row & 7)];  // w1[o][c]
    wp[192 + e] = v;
  }

  if (threadIdx.x < 16) {
    const int col = threadIdx.x, i = col >> 3, o = col & 7;
    wp[448 + col] = sp[i][160 + o];   // b1
    wp[464 + col] = sp[i][144 + o];   // w2 (packed [inst0|inst1])
  }
  if (threadIdx.x < 2) wp[480 + threadIdx.x] = sp[threadIdx.x][168];  // b2
}

// Half-wave exchange producing two merged registers from (a,b):
//   result.x = {a.r0 (lanes 0-15), b.r0 (lanes 16-31)}
//   result.y = {a.r1 (lanes 0-15), b.r1 (lanes 16-31)}
static __device__ __forceinline__ v2f half_swap(float a, float b) {
  v2f out;
#if __has_builtin(__builtin_amdgcn_permlane16_swap)
  typedef unsigned v2u __attribute__((ext_vector_type(2)));
  const v2u r = __builtin_amdgcn_permlane16_swap(__float_as_uint(a),
                                                 __float_as_uint(b),
                                                 false, false);
  out.x = __uint_as_float(r.x);
  out.y = __uint_as_float(r.y);
#else
  const float sa = __shfl_xor(a, 16, 32);   // {a.r1, a.r0}
  const float sb = __shfl_xor(b, 16, 32);   // {b.r1, b.r0}
  const bool upper = (threadIdx.x & 16) != 0;
  out.x = upper ? sb : a;
  out.y = upper ? b : sa;
#endif
  return out;
}

// Build the 4 B-operand fragments (K in {0..15}) from a D-layout 16x16 matrix.
static __device__ __forceinline__ void d_to_bfrags(const v8f& x, v2f bf[4]) {
  const v2f p0 = half_swap(x[0], x[2]);  // -> bf[0].x, bf[2].x
  const v2f p1 = half_swap(x[1], x[3]);  // -> bf[0].y, bf[2].y
  const v2f p2 = half_swap(x[4], x[6]);  // -> bf[1].x, bf[3].x
  const v2f p3 = half_swap(x[5], x[7]);  // -> bf[1].y, bf[3].y
  bf[0].x = p0.x; bf[2].x = p0.y;
  bf[0].y = p1.x; bf[2].y = p1.y;
  bf[1].x = p2.x; bf[3].x = p2.y;
  bf[1].y = p3.x; bf[3].y = p3.y;
}

__global__ void __launch_bounds__(256, 2)
main_kernel(const float* __restrict__ seg,
            const float* __restrict__ target,
            const float* __restrict__ ws,
            float* __restrict__ partials) {
  __shared__ float red[3][8];

  const int pair = blockIdx.x;
  const int n = pair >> 4;
  const int ip = pair & 15;
  const int m0 = n * KOBJ + ip * 2;

  const int lane = threadIdx.x & 31;
  const int wv = threadIdx.x >> 5;
  const int col = lane & 15;       // A: M-channel lane / B: N-pixel lane
  const int hi = lane >> 4;        // lane half

  const float* wp = ws + (size_t)pair * PAIR_STRIDE;

  // per-wave constant A-operand fragments (weights)
  v2f w0f[3], w1f[4], w2f[4];
#pragma unroll
  for (int kg = 0; kg < 3; ++kg)
    w0f[kg] = *(const v2f*)(wp + ((kg * 2 + hi) * 16 + col) * 2);
#pragma unroll
  for (int kg = 0; kg < 4; ++kg)
    w1f[kg] = *(const v2f*)(wp + 192 + ((kg * 2 + hi) * 16 + col) * 2);
  // A2: row 0 = w2 of instance 0 (K 0..7), row 1 = w2 of instance 1 (K 8..15)
#pragma unroll
  for (int kg = 0; kg < 4; ++kg) {
    const int r0 = 4 * kg + 2 * hi;
    w2f[kg].x = (col == (r0 >> 3)) ? wp[464 + r0] : 0.0f;
    w2f[kg].y = (col == ((r0 + 1) >> 3)) ? wp[464 + r0 + 1] : 0.0f;
  }
  // C-inits: biases per output channel (channel index = v + 8*hi)
  v8f c1vec, c2vec;
#pragma unroll
  for (int v = 0; v < 8; ++v) {
    c1vec[v] = wp[448 + v + 8 * hi];
    c2vec[v] = wp[480 + ((v + 8 * hi) & 1)];
  }

  const float* segn = seg + (size_t)n * CFEAT * HWSZ;
  const float* tgt0 = target + (size_t)m0 * HWSZ;

  float ai = 0.0f, as = 0.0f, at = 0.0f;

  const int t0 = wv * 128;
  for (int t = t0; t < t0 + 128; ++t) {
    const int y = t >> 3;
    const int x0 = (t & 7) << 4;
    const int sp_off = y * WW + x0 + col;

    // B-operand fragments: 12 features x 16 pixels [seg(8), x, y, 1, 0]
    v2f f0, f1, f2;
    f0.x = segn[(2 * hi + 0) * HWSZ + sp_off];
    f0.y = segn[(2 * hi + 1) * HWSZ + sp_off];
    f1.x = segn[(2 * hi + 4) * HWSZ + sp_off];
    f1.y = segn[(2 * hi + 5) * HWSZ + sp_off];
    f2.x = hi ? 1.0f : (float)(x0 + col);
    f2.y = hi ? 0.0f : (float)y;

    // ---- layer 0: 16ch x 12 @ 12 x 16px ----
    v8f acc = {0.f, 0.f, 0.f, 0.f, 0.f, 0.f, 0.f, 0.f};
    acc = __builtin_amdgcn_wmma_f32_16x16x4_f32(false, w0f[0], false, f0,
                                                (short)0, acc, false, false);
    acc = __builtin_amdgcn_wmma_f32_16x16x4_f32(false, w0f[1], false, f1,
                                                (short)0, acc, false, false);
    acc = __builtin_amdgcn_wmma_f32_16x16x4_f32(false, w0f[2], false, f2,
                                                (short)0, acc, false, false);

    // relu + half-wave swap: D layout -> B-operand layout (no LDS!)
#pragma unroll
    for (int v = 0; v < 8; ++v) acc[v] = __builtin_fmaxf(acc[v], 0.0f);
    v2f x1f[4];
    d_to_bfrags(acc, x1f);

    // ---- layer 1: 16ch x 16 @ 16 x 16px, block-diagonal, bias via C ----
    v8f acc1 = c1vec;
#pragma unroll
    for (int kg = 0; kg < 4; ++kg)
      acc1 = __builtin_amdgcn_wmma_f32_16x16x4_f32(false, w1f[kg], false,
                                                   x1f[kg], (short)0, acc1,
                                                   false, false);

#pragma unroll
    for (int v = 0; v < 8; ++v) acc1[v] = __builtin_fmaxf(acc1[v], 0.0f);
    v2f x2f[4];
    d_to_bfrags(acc1, x2f);

    // ---- layer 2: rows 0/1 of D = logits of the two instances ----
    v8f acc2 = c2vec;
#pragma unroll
    for (int kg = 0; kg < 4; ++kg)
      acc2 = __builtin_amdgcn_wmma_f32_16x16x4_f32(false, w2f[kg], false,
                                                   x2f[kg], (short)0, acc2,
                                                   false, false);

    // dice partials: lanes 0..15 hold pixel x0+col; D row0=inst0, row1=inst1
    if (hi == 0) {
      const size_t po = (size_t)y * WW + x0 + col;
      const float t0v = tgt0[po];
      const float t1v = tgt0[HWSZ + po];
      const float s0 = 1.0f / (1.0f + __expf(-acc2[0]));
      const float s1 = 1.0f / (1.0f + __expf(-acc2[1]));
      ai += s0 * t0v + s1 * t1v;
      as += s0 * s0 + s1 * s1;
      at += t0v * t0v + t1v * t1v;
    }
  }

  // deterministic in-block reduction
#pragma unroll
  for (int d = 16; d >= 1; d >>= 1) {
    ai += __shfl_xor(ai, d, 32);
    as += __shfl_xor(as, d, 32);
    at += __shfl_xor(at, d, 32);
  }
  if (lane == 0) { red[0][wv] = ai; red[1][wv] = as; red[2][wv] = at; }
  __syncthreads();
  if (threadIdx.x == 0) {
    float A = 0.f, B = 0.f, Cv = 0.f;
#pragma unroll
    for (int i = 0; i < 8; ++i) { A += red[0][i]; B += red[1][i]; Cv += red[2][i]; }
    partials[blockIdx.x * 3 + 0] = A;
    partials[blockIdx.x * 3 + 1] = B;
    partials[blockIdx.x * 3 + 2] = Cv;
  }
}

__global__ void fin_kernel(const float* __restrict__ partials,
                           float* __restrict__ out) {
  __shared__ float sa[128], sb[128], sc[128];
  const int t = threadIdx.x;
  sa[t] = partials[t * 3 + 0];
  sb[t] = partials[t * 3 + 1];
  sc[t] = partials[t * 3 + 2];
  __syncthreads();
  for (int d = 64; d >= 1; d >>= 1) {
    if (t < d) { sa[t] += sa[t + d]; sb[t] += sb[t + d]; sc[t] += sc[t + d]; }
    __syncthreads();
  }
  if (t == 0) {
    const float inter = sa[0], ss = sb[0], st = sc[0];
    out[0] = (1.0f - (2.0f * inter + 1.0f) / (ss + st + 1.0f)) / (float)NIMG;
  }
}

extern "C" void kernel_launch(void* const* d_in, const int* in_sizes, int n_in,
                              void* d_out, int out_size, void* d_ws, size_t ws_size,
                              hipStream_t stream) {
  (void)in_sizes; (void)n_in; (void)out_size; (void)ws_size;
  const float* seg    = (const float*)d_in[0];   // (8,8,128,128)
  const float* cw     = (const float*)d_in[1];   // (8,169,128,128)
  // d_in[2] = mask (unused by the reference loss)
  const int*   ind    = (const int*)d_in[3];     // (8,32)
  const float* target = (const float*)d_in[4];   // (8,32,128,128)
  // d_in[5] = nums (static, unused)

  float* wsf = (float*)d_ws;
  float* partials = wsf + ACC_BASE;

  prep_kernel<<<NPAIR, 256, 0, stream>>>(cw, ind, wsf);
  main_kernel<<<NPAIR, 256, 0, stream>>>(seg, target, wsf, partials);
  fin_kernel<<<1, 128, 0, stream>>>(partials, (float*)d_out);
}